// SetfnTopologyLayer_3006477107666
// MI455X (gfx1250) — compile-verified
//
#include <hip/hip_runtime.h>
#include <math.h>

typedef __attribute__((ext_vector_type(2))) float v2f;
typedef __attribute__((ext_vector_type(8))) float v8f;

#define NN 65536
#define FF 256
#define EE 524288
#define BB 512
#define NPG 128
#define NFt 16
#define HH 256
#define DD0 128

static __device__ __forceinline__ v8f wmma4(v2f a, v2f b, v8f c) {
  // D = A(16x4,f32) * B(4x16,f32) + C(16x16,f32)
  return __builtin_amdgcn_wmma_f32_16x16x4_f32(false, a, false, b, (short)0, c, false, false);
}

// ---------------- K0: zero BN accumulators ----------------
__global__ void k0_zero(float* gsum, float* gsumsq) {
  gsum[threadIdx.x] = 0.f;
  gsumsq[threadIdx.x] = 0.f;
}

// ---------------- K1: fv = relu(x@W1+b1)@W2+b2 ----------------
// 64-row tile per block; 8 waves = 4 M-tiles x 2 N-halves of the hidden layer.
// W1 streamed through LDS in paired-K 32-row panels; x tile reused for hidden.
__global__ void k1_filt(const float* __restrict__ x,
                        const float* __restrict__ w1, const float* __restrict__ b1,
                        const float* __restrict__ w2, const float* __restrict__ b2,
                        float* __restrict__ fv) {
  __shared__ float AH[64 * 256];   // x tile (64x256), later hidden tile
  __shared__ v2f   Wsp[16 * 256];  // W1 K-panel: {W1[2k,c],W1[2k+1,c]} pairs
  const int tid = threadIdx.x;
  const int wave = tid >> 5, lane = tid & 31;
  const int lr = lane & 15, lh = lane >> 4;
  const int wm = wave >> 1;        // 0..3 : M sub-tile
  const int wn = wave & 1;         // 0..1 : N half (128 cols)
  const int row0 = blockIdx.x * 64;

  // stage x tile (64x256) into LDS, coalesced float4 (16 per thread)
  {
    const float4* src = (const float4*)(x + (size_t)row0 * 256);
    float4* dst = (float4*)AH;
    #pragma unroll
    for (int i = 0; i < 16; ++i) dst[tid + i * 256] = src[tid + i * 256];
  }

  v8f c[8];
  #pragma unroll
  for (int nt = 0; nt < 8; ++nt) c[nt] = (v8f){0.f,0.f,0.f,0.f,0.f,0.f,0.f,0.f};

  const int arow = (wm * 16 + lr) * 256;
  for (int p = 0; p < 8; ++p) {            // K panels of 32
    // stage W1 rows [32p, 32p+32) x 256 cols, paired-K layout
    #pragma unroll
    for (int i = 0; i < 4; ++i) {
      const int u = tid + i * 256;         // 1024 units: 16 rowpairs x 64 col4
      const int rp = u >> 6, c4 = (u & 63) * 4;
      const float4 g0 = *(const float4*)&w1[(size_t)(p * 32 + 2 * rp) * 256 + c4];
      const float4 g1 = *(const float4*)&w1[(size_t)(p * 32 + 2 * rp + 1) * 256 + c4];
      Wsp[rp * 256 + c4 + 0] = (v2f){g0.x, g1.x};
      Wsp[rp * 256 + c4 + 1] = (v2f){g0.y, g1.y};
      Wsp[rp * 256 + c4 + 2] = (v2f){g0.z, g1.z};
      Wsp[rp * 256 + c4 + 3] = (v2f){g0.w, g1.w};
    }
    __syncthreads();
    if (p < 7) // prefetch next panel: one 128B line per thread -> 32KB
      __builtin_prefetch(w1 + (size_t)(p + 1) * 32 * 256 + tid * 32, 0, 1);

    for (int kk = 0; kk < 32; kk += 4) {
      const int ka = kk + 2 * lh;
      const int kp = (kk >> 1) + lh;
      const v2f a = *(const v2f*)&AH[arow + p * 32 + ka];
      #pragma unroll
      for (int nt = 0; nt < 8; ++nt) {
        const int col = wn * 128 + nt * 16 + lr;
        c[nt] = wmma4(a, Wsp[kp * 256 + col], c[nt]);
      }
    }
    __syncthreads();                        // protect Wsp (and final AH reads)
  }

  // epilogue: hidden = relu(acc + b1) overwrites AH (all AH reads done)
  #pragma unroll
  for (int nt = 0; nt < 8; ++nt) {
    const int col = wn * 128 + nt * 16 + lr;
    const float bias = b1[col];
    #pragma unroll
    for (int v = 0; v < 8; ++v) {
      const int row = wm * 16 + v + 8 * lh;
      const float h = c[nt][v] + bias;
      AH[row * 256 + col] = h > 0.f ? h : 0.f;
    }
  }
  __syncthreads();

  // fv tile (64x16) = hidden @ W2 + b2 ; 4 waves, one 16-row M-tile each (K=256)
  if (wave < 4) {
    v8f cc = (v8f){0.f,0.f,0.f,0.f,0.f,0.f,0.f,0.f};
    const int hrow = (wave * 16 + lr) * 256;
    for (int k0 = 0; k0 < 256; k0 += 4) {
      const int ka = k0 + 2 * lh;
      const v2f a = *(const v2f*)&AH[hrow + ka];
      v2f b; b.x = w2[ka * 16 + lr]; b.y = w2[(ka + 1) * 16 + lr];
      cc = wmma4(a, b, cc);
    }
    const float bias = b2[lr];
    #pragma unroll
    for (int v = 0; v < 8; ++v) {
      const int row = wave * 16 + v + 8 * lh;
      fv[(size_t)(row0 + row) * 16 + lr] = cc[v] + bias;
    }
  }
}

// ---------------- K2: per-graph persistence branch -> x1 (B x 256) --------
__global__ void k2_topo(const float* __restrict__ fv,
                        const int* __restrict__ edge_index,
                        const int* __restrict__ random_edges,
                        const float* __restrict__ fn1_w,
                        const float* __restrict__ fn1_b,
                        const float* __restrict__ dsd1_lw,
                        float* __restrict__ x1) {
  __shared__ int   eidx[16];
  __shared__ float val[16];
  __shared__ int   canon[16];
  __shared__ int   maskv[16];
  __shared__ float c1s;
  __shared__ float xm1[128];
  const int b = blockIdx.x, tid = threadIdx.x;

  if (tid < 16) {
    const int e = random_edges[b * 16 + tid];
    eidx[tid] = e;
    const int s = edge_index[e];
    const int d = edge_index[EE + e];
    const float a = fv[(size_t)s * 16 + tid], bb = fv[(size_t)d * 16 + tid];
    val[tid] = a > bb ? a : bb;
  }
  __syncthreads();
  if (tid == 0) {
    for (int j = 0; j < 16; ++j) {
      int cj = j;
      for (int jp = 0; jp < j; ++jp) if (eidx[jp] == eidx[j]) { cj = jp; break; }
      canon[j] = cj;
    }
    int cnt = 0;
    for (int j = 0; j < 16; ++j) {
      int m = 0;
      if (canon[j] == j) {
        for (int jp = j; jp < 16; ++jp)
          if (canon[jp] == j && val[jp] != 0.f) m = 1;
        if (m) cnt++;
      }
      maskv[j] = m;
    }
    c1s = (float)(cnt >= 1 ? cnt : 1);
  }
  __syncthreads();
  if (tid < 128) {
    float s1 = 0.f;
    for (int j = 0; j < 16; ++j) {
      if (canon[j] == j && maskv[j]) {
        float acc = fn1_b[tid];
        for (int jp = j; jp < 16; ++jp)
          if (canon[jp] == j) acc += val[jp] * fn1_w[(2 * jp + 1) * 128 + tid];
        s1 += acc > 0.f ? acc : 0.f;
      }
    }
    xm1[tid] = s1 / c1s;
  }
  __syncthreads();
  if (tid < 128) {
    for (int h = 0; h < 2; ++h) {
      const int f = tid + h * 128;
      float s = 0.f;
      for (int d = 0; d < 128; ++d) s += xm1[d] * dsd1_lw[d * 256 + f];
      x1[b * 256 + f] = s;
    }
  }
}

// ---------------- K3: per-graph fused deepset stack -> r, BN partial sums --
// H0 updated in place (GEMM2 keeps tiles in registers); gw weights staged
// once per block through a 16KB paired-K LDS panel.
__global__ void k3_deepset(const float* __restrict__ fv,
                           const float* __restrict__ fn0_w, const float* __restrict__ fn0_b,
                           const float* __restrict__ ds1_gw, const float* __restrict__ ds1_gb,
                           const float* __restrict__ ds1_lw,
                           const float* __restrict__ ds2_gw, const float* __restrict__ ds2_gb,
                           const float* __restrict__ ds2_lw,
                           const float* __restrict__ x1,
                           float* __restrict__ rbuf,
                           float* __restrict__ gsum, float* __restrict__ gsumsq) {
  __shared__ float H0[128 * 128];
  __shared__ v2f   Wp[16 * 128];    // 32 K-rows x 128 cols, paired-K
  __shared__ float xmv[128];
  __shared__ float tv[256];
  __shared__ float colsum[256];
  __shared__ float colsq[256];

  const int g = blockIdx.x, tid = threadIdx.x;
  const int wave = tid >> 5, lane = tid & 31;
  const int lr = lane & 15, lh = lane >> 4;
  const int row0g = g * 128;
  const int mrow = wave * 16;

  colsum[tid] = 0.f; colsq[tid] = 0.f;

  // GEMM1: H0 = relu(fv @ W' + fn0_b), W'[k] = fn0_w[2k]+fn0_w[2k+1], K=16
  {
    v8f acc[8];
    for (int nt = 0; nt < 8; ++nt) {
      v8f c = {0.f,0.f,0.f,0.f,0.f,0.f,0.f,0.f};
      for (int k0 = 0; k0 < 16; k0 += 4) {
        const int ka = k0 + 2 * lh;
        const int col = nt * 16 + lr;
        const v2f a = *(const v2f*)&fv[(size_t)(row0g + mrow + lr) * 16 + ka];
        v2f b;
        b.x = fn0_w[(2 * ka) * 128 + col]     + fn0_w[(2 * ka + 1) * 128 + col];
        b.y = fn0_w[(2 * ka + 2) * 128 + col] + fn0_w[(2 * ka + 3) * 128 + col];
        c = wmma4(a, b, c);
      }
      acc[nt] = c;
    }
    for (int nt = 0; nt < 8; ++nt) {
      const int col = nt * 16 + lr;
      const float bias = fn0_b[col];
      #pragma unroll
      for (int v = 0; v < 8; ++v) {
        const float h = acc[nt][v] + bias;
        H0[(mrow + v + 8 * lh) * 128 + col] = h > 0.f ? h : 0.f;
      }
    }
  }
  __syncthreads();
  if (tid < 128) {                        // column mean of H0
    float s = 0.f;
    for (int r = 0; r < 128; ++r) s += H0[r * 128 + tid];
    xmv[tid] = s * (1.f / 128.f);
  }
  __syncthreads();
  if (tid < 128) {                        // tv = ds1_gb - xm0 @ ds1_lw
    float s = 0.f;
    for (int d = 0; d < 128; ++d) s += xmv[d] * ds1_lw[d * 128 + tid];
    tv[tid] = ds1_gb[tid] - s;
  }
  __syncthreads();

  v8f acc[8];

  // GEMM2: H0 <- relu(H0 @ ds1_gw + tv), K=128, in place (reg-buffered)
  #pragma unroll
  for (int nt = 0; nt < 8; ++nt) acc[nt] = (v8f){0.f,0.f,0.f,0.f,0.f,0.f,0.f,0.f};
  for (int p = 0; p < 4; ++p) {
    #pragma unroll
    for (int i = 0; i < 2; ++i) {          // 512 units: 16 rowpairs x 32 col4
      const int u = tid + i * 256;
      const int rp = u >> 5, c4 = (u & 31) * 4;
      const float4 g0 = *(const float4*)&ds1_gw[(p * 32 + 2 * rp) * 128 + c4];
      const float4 g1 = *(const float4*)&ds1_gw[(p * 32 + 2 * rp + 1) * 128 + c4];
      Wp[rp * 128 + c4 + 0] = (v2f){g0.x, g1.x};
      Wp[rp * 128 + c4 + 1] = (v2f){g0.y, g1.y};
      Wp[rp * 128 + c4 + 2] = (v2f){g0.z, g1.z};
      Wp[rp * 128 + c4 + 3] = (v2f){g0.w, g1.w};
    }
    __syncthreads();
    for (int kk = 0; kk < 32; kk += 4) {
      const int ka = p * 32 + kk + 2 * lh;
      const int kp = (kk >> 1) + lh;
      const v2f a = *(const v2f*)&H0[(mrow + lr) * 128 + ka];
      #pragma unroll
      for (int nt = 0; nt < 8; ++nt)
        acc[nt] = wmma4(a, Wp[kp * 128 + nt * 16 + lr], acc[nt]);
    }
    __syncthreads();
  }
  #pragma unroll
  for (int nt = 0; nt < 8; ++nt) {        // in-place epilogue (own rows only)
    const int col = nt * 16 + lr;
    const float add = tv[col];
    #pragma unroll
    for (int v = 0; v < 8; ++v) {
      const float h = acc[nt][v] + add;
      H0[(mrow + v + 8 * lh) * 128 + col] = h > 0.f ? h : 0.f;
    }
  }
  __syncthreads();
  if (tid < 128) {                        // column mean of updated H0
    float s = 0.f;
    for (int r = 0; r < 128; ++r) s += H0[r * 128 + tid];
    xmv[tid] = s * (1.f / 128.f);
  }
  __syncthreads();
  {                                       // tv = ds2_gb - xm @ ds2_lw + x1[g]
    float s = 0.f;
    for (int d = 0; d < 128; ++d) s += xmv[d] * ds2_lw[d * 256 + tid];
    tv[tid] = ds2_gb[tid] - s + x1[g * 256 + tid];
  }
  __syncthreads();

  // GEMM3: r = relu(H0 @ ds2_gw + tv) -> rbuf ; two 128-col halves
  for (int half = 0; half < 2; ++half) {
    #pragma unroll
    for (int nt = 0; nt < 8; ++nt) acc[nt] = (v8f){0.f,0.f,0.f,0.f,0.f,0.f,0.f,0.f};
    for (int p = 0; p < 4; ++p) {
      #pragma unroll
      for (int i = 0; i < 2; ++i) {        // stage 32x128 slice of ds2_gw
        const int u = tid + i * 256;
        const int rp = u >> 5, c4 = (u & 31) * 4;
        const float4 g0 = *(const float4*)&ds2_gw[(p * 32 + 2 * rp) * 256 + half * 128 + c4];
        const float4 g1 = *(const float4*)&ds2_gw[(p * 32 + 2 * rp + 1) * 256 + half * 128 + c4];
        Wp[rp * 128 + c4 + 0] = (v2f){g0.x, g1.x};
        Wp[rp * 128 + c4 + 1] = (v2f){g0.y, g1.y};
        Wp[rp * 128 + c4 + 2] = (v2f){g0.z, g1.z};
        Wp[rp * 128 + c4 + 3] = (v2f){g0.w, g1.w};
      }
      __syncthreads();
      for (int kk = 0; kk < 32; kk += 4) {
        const int ka = p * 32 + kk + 2 * lh;
        const int kp = (kk >> 1) + lh;
        const v2f a = *(const v2f*)&H0[(mrow + lr) * 128 + ka];
        #pragma unroll
        for (int nt = 0; nt < 8; ++nt)
          acc[nt] = wmma4(a, Wp[kp * 128 + nt * 16 + lr], acc[nt]);
      }
      __syncthreads();
    }
    #pragma unroll
    for (int nt = 0; nt < 8; ++nt) {
      const int colg = half * 128 + nt * 16 + lr;
      const float add = tv[colg];
      float psum = 0.f, psq = 0.f;
      #pragma unroll
      for (int v = 0; v < 8; ++v) {
        const int grow = row0g + mrow + v + 8 * lh;
        float r = acc[nt][v] + add;
        r = r > 0.f ? r : 0.f;
        rbuf[(size_t)grow * 256 + colg] = r;
        psum += r; psq += r * r;
      }
      atomicAdd(&colsum[colg], psum);
      atomicAdd(&colsq[colg], psq);
    }
  }
  __syncthreads();
  atomicAdd(&gsum[tid], colsum[tid]);
  atomicAdd(&gsumsq[tid], colsq[tid]);
}

// ---------------- K5: fold BN stats into per-feature scale/shift ----------
__global__ void k5_bnprep(const float* __restrict__ gsum, const float* __restrict__ gsumsq,
                          const float* __restrict__ gamma, const float* __restrict__ beta,
                          float* __restrict__ scale, float* __restrict__ shift) {
  const int f = threadIdx.x;
  const float mu  = gsum[f]   * (1.f / (float)NN);
  const float var = gsumsq[f] * (1.f / (float)NN) - mu * mu;
  const float inv = rsqrtf(var + 1e-5f);
  const float sc = gamma[f] * inv;
  scale[f] = sc;
  shift[f] = beta[f] - mu * sc;
}

// ---------------- K4: out = x + r*scale + shift (float4 streaming) --------
__global__ void k4_bn(const float* __restrict__ x, const float* __restrict__ r,
                      const float* __restrict__ scale, const float* __restrict__ shift,
                      float* __restrict__ out) {
  const size_t i4 = (size_t)blockIdx.x * 256 + threadIdx.x;
  const int f0 = (int)((i4 * 4) & 255);
  const float4 xv = ((const float4*)x)[i4];
  const float4 rv = ((const float4*)r)[i4];
  float4 o;
  o.x = xv.x + rv.x * scale[f0 + 0] + shift[f0 + 0];
  o.y = xv.y + rv.y * scale[f0 + 1] + shift[f0 + 1];
  o.z = xv.z + rv.z * scale[f0 + 2] + shift[f0 + 2];
  o.w = xv.w + rv.w * scale[f0 + 3] + shift[f0 + 3];
  ((float4*)out)[i4] = o;
}

extern "C" void kernel_launch(void* const* d_in, const int* in_sizes, int n_in,
                              void* d_out, int out_size, void* d_ws, size_t ws_size,
                              hipStream_t stream) {
  const float* x          = (const float*)d_in[0];
  const int*   edge_index = (const int*)d_in[1];
  // d_in[2] batch, d_in[3] edge_slices: structure is implied by constants
  const int*   random_edges = (const int*)d_in[4];
  const float* filt_w1 = (const float*)d_in[5];
  const float* filt_b1 = (const float*)d_in[6];
  const float* filt_w2 = (const float*)d_in[7];
  const float* filt_b2 = (const float*)d_in[8];
  const float* fn0_w   = (const float*)d_in[9];
  const float* fn0_b   = (const float*)d_in[10];
  const float* ds1_gw  = (const float*)d_in[11];
  const float* ds1_gb  = (const float*)d_in[12];
  const float* ds1_lw  = (const float*)d_in[13];
  const float* ds2_gw  = (const float*)d_in[14];
  const float* ds2_gb  = (const float*)d_in[15];
  const float* ds2_lw  = (const float*)d_in[16];
  const float* fn1_w   = (const float*)d_in[17];
  const float* fn1_b   = (const float*)d_in[18];
  const float* dsd1_lw = (const float*)d_in[19];
  const float* bn_gamma = (const float*)d_in[20];
  const float* bn_beta  = (const float*)d_in[21];

  float* fv     = (float*)d_ws;                 // N*16
  float* x1     = fv + (size_t)NN * 16;         // B*256
  float* gsum   = x1 + (size_t)BB * 256;        // 256
  float* gsumsq = gsum + 256;                   // 256
  float* scale  = gsumsq + 256;                 // 256
  float* shift  = scale + 256;                  // 256
  float* rbuf   = shift + 256;                  // N*256

  k0_zero<<<1, 256, 0, stream>>>(gsum, gsumsq);
  k1_filt<<<NN / 64, 256, 0, stream>>>(x, filt_w1, filt_b1, filt_w2, filt_b2, fv);
  k2_topo<<<BB, 128, 0, stream>>>(fv, edge_index, random_edges, fn1_w, fn1_b, dsd1_lw, x1);
  k3_deepset<<<BB, 256, 0, stream>>>(fv, fn0_w, fn0_b, ds1_gw, ds1_gb, ds1_lw,
                                     ds2_gw, ds2_gb, ds2_lw, x1, rbuf, gsum, gsumsq);
  k5_bnprep<<<1, 256, 0, stream>>>(gsum, gsumsq, bn_gamma, bn_beta, scale, shift);
  k4_bn<<<(int)(((size_t)NN * 256) / 4 / 256), 256, 0, stream>>>(x, rbuf, scale, shift,
                                                                 (float*)d_out);
}